// RecurrentGCN_EGCNO_80504866996300
// MI455X (gfx1250) — compile-verified
//
#include <hip/hip_runtime.h>
#include <hip/hip_bf16.h>

#define IN_C 128

typedef float v2f __attribute__((ext_vector_type(2)));
typedef float v8f __attribute__((ext_vector_type(8)));
typedef int   v4i __attribute__((ext_vector_type(4)));

// ---------------------------------------------------------------------------
// Async global -> LDS 16B copy (CDNA5 GLOBAL_LOAD_ASYNC_TO_LDS_B128,
// tracked with ASYNCcnt). Builtin signature (from hipcc diagnostic):
//   (as1 v4i*, as3 v4i*, i32 offset, i32 cpol)
// ---------------------------------------------------------------------------
__device__ __forceinline__ void async_copy_b128_g2l(const float* gsrc, float* ldst) {
  __builtin_amdgcn_global_load_async_to_lds_b128(
      (__attribute__((address_space(1))) v4i*)gsrc,
      (__attribute__((address_space(3))) v4i*)ldst, 0, 0);
}

__device__ __forceinline__ void wait_async_all() {
#if defined(__has_builtin) && __has_builtin(__builtin_amdgcn_s_wait_asynccnt)
  __builtin_amdgcn_s_wait_asynccnt(0);
#else
  asm volatile("s_wait_asynccnt 0x0" ::: "memory");
#endif
}

// ---------------------------------------------------------------------------
// K1: EvolveGCNO GRU step: W = (1-z)*tanh(i_n + r*h_n) + z*W0
// gi = W0 @ W_ih^T + b_ih ; gh = W0 @ W_hh^T + b_hh  (gate order r,z,n)
// One block per row i of W0; thread j computes column j. 25 MFLOP total.
// ---------------------------------------------------------------------------
__global__ void __launch_bounds__(IN_C) evolve_W_kernel(
    const float* __restrict__ W0, const float* __restrict__ Wih,
    const float* __restrict__ Whh, const float* __restrict__ bih,
    const float* __restrict__ bhh, float* __restrict__ W) {
  __shared__ float w0row[IN_C];
  const int i = blockIdx.x;   // row of W0 (0..127)
  const int j = threadIdx.x;  // col (0..127)
  w0row[j] = W0[i * IN_C + j];
  __syncthreads();

  float gi_r = bih[j], gi_z = bih[j + IN_C], gi_n = bih[j + 2 * IN_C];
  float gh_r = bhh[j], gh_z = bhh[j + IN_C], gh_n = bhh[j + 2 * IN_C];
  const float* wih_r = Wih + (size_t)j * IN_C;
  const float* wih_z = Wih + (size_t)(j + IN_C) * IN_C;
  const float* wih_n = Wih + (size_t)(j + 2 * IN_C) * IN_C;
  const float* whh_r = Whh + (size_t)j * IN_C;
  const float* whh_z = Whh + (size_t)(j + IN_C) * IN_C;
  const float* whh_n = Whh + (size_t)(j + 2 * IN_C) * IN_C;
#pragma unroll 4
  for (int k = 0; k < IN_C; ++k) {
    const float a = w0row[k];
    gi_r = fmaf(a, wih_r[k], gi_r);
    gi_z = fmaf(a, wih_z[k], gi_z);
    gi_n = fmaf(a, wih_n[k], gi_n);
    gh_r = fmaf(a, whh_r[k], gh_r);
    gh_z = fmaf(a, whh_z[k], gh_z);
    gh_n = fmaf(a, whh_n[k], gh_n);
  }
  const float r = 1.0f / (1.0f + expf(-(gi_r + gh_r)));
  const float z = 1.0f / (1.0f + expf(-(gi_z + gh_z)));
  const float cand = tanhf(gi_n + r * gh_n);
  W[i * IN_C + j] = (1.0f - z) * cand + z * w0row[j];
}

// ---------------------------------------------------------------------------
// K2: zero degree array
// ---------------------------------------------------------------------------
__global__ void zero_f32_kernel(float* __restrict__ p, int n) {
  const int i = blockIdx.x * 256 + threadIdx.x;
  if (i < n) p[i] = 0.0f;
}

// ---------------------------------------------------------------------------
// K3: deg[col] += w  (edge part of segment_sum)
// ---------------------------------------------------------------------------
__global__ void deg_accum_kernel(const int* __restrict__ ei,
                                 const float* __restrict__ ew,
                                 float* __restrict__ deg, int ne) {
  const int e = blockIdx.x * 256 + threadIdx.x;
  if (e < ne) atomicAdd(deg + ei[ne + e], ew[e]);
}

// ---------------------------------------------------------------------------
// K4: deg -> dinv in place: d = deg + 1 (self-loop); dinv = d>0 ? rsqrt(d) : 0
// ---------------------------------------------------------------------------
__global__ void dinv_kernel(float* __restrict__ deg, int n) {
  const int i = blockIdx.x * 256 + threadIdx.x;
  if (i < n) {
    const float d = deg[i] + 1.0f;
    deg[i] = (d > 0.0f) ? rsqrtf(fmaxf(d, 1e-30f)) : 0.0f;
  }
}

// ---------------------------------------------------------------------------
// K5: h = x @ W  via V_WMMA_F32_16X16X4_F32 (fp32 in, fp32 acc).
// Block = 256 threads = 8 waves; each wave owns 16 rows of x (16x128 A strip
// held in 64 VGPRs) and produces 16x128 of h. Column tiles are processed in
// PAIRS -> two independent accumulator chains keep the matrix pipe busy.
// W (64 KB) is DMA'd into LDS with GLOBAL_LOAD_ASYNC_TO_LDS_B128.
//
// f32 WMMA VGPR layouts (ISA 7.12.2):
//   A 16x4 : lanes 0-15 -> {K0,K1} in V0,V1 ; lanes 16-31 -> {K2,K3}
//   B 4x16 : lanes hold N; lane-group selects K pair (mirror of A)
//   C/D    : VGPR v -> row v (lanes 0-15) / row v+8 (lanes 16-31)
// ---------------------------------------------------------------------------
__global__ void __launch_bounds__(256) gemm_xW_kernel(
    const float* __restrict__ x, const float* __restrict__ W,
    float* __restrict__ h, int n_nodes) {
  extern __shared__ float Ws[];  // IN_C*IN_C floats = 64 KB

  // Async DMA W -> LDS: 16 iters x 256 threads x 16B = 64 KB (ASYNCcnt).
#pragma unroll
  for (int it = 0; it < 16; ++it) {
    const int off4 = (it * 256 + threadIdx.x) * 4;  // float offset, 16B granules
    async_copy_b128_g2l(W + off4, Ws + off4);
  }
  wait_async_all();   // this wave's DMAs complete
  __syncthreads();    // all waves' DMAs complete

  const int wave = threadIdx.x >> 5;
  const int lane = threadIdx.x & 31;
  const int row0 = (blockIdx.x * 8 + wave) * 16;  // first of 16 rows
  const int mrow = lane & 15;
  const int khalf = (lane >> 4) * 2;  // 0 for lanes 0-15, 2 for lanes 16-31
  const int col = lane & 15;

  // A strip: clamp OOB rows (their outputs are never stored).
  int grow = row0 + mrow;
  if (grow >= n_nodes) grow = n_nodes - 1;
  const float* xrow = x + (size_t)grow * IN_C;

  v2f A[32];
#pragma unroll
  for (int ks = 0; ks < 32; ++ks) {
    const float2 t = *(const float2*)(xrow + ks * 4 + khalf);
    A[ks] = (v2f){t.x, t.y};
  }

  const int rbase = row0 + ((lane >> 4) ? 8 : 0);
  for (int t = 0; t < 8; t += 2) {  // two 16-col tiles per iteration
    v8f acc0 = (v8f)0.0f;
    v8f acc1 = (v8f)0.0f;
#pragma unroll
    for (int ks = 0; ks < 32; ++ks) {
      const int k = ks * 4 + khalf;
      v2f b0, b1;
      b0.x = Ws[k * IN_C + t * 16 + col];
      b0.y = Ws[(k + 1) * IN_C + t * 16 + col];
      b1.x = Ws[k * IN_C + (t + 1) * 16 + col];
      b1.y = Ws[(k + 1) * IN_C + (t + 1) * 16 + col];
      acc0 = __builtin_amdgcn_wmma_f32_16x16x4_f32(
          false, A[ks], false, b0, (short)0, acc0, false, false);
      acc1 = __builtin_amdgcn_wmma_f32_16x16x4_f32(
          false, A[ks], false, b1, (short)0, acc1, false, false);
    }
#pragma unroll
    for (int v = 0; v < 8; ++v) {
      const int r = rbase + v;
      if (r < n_nodes) {
        h[(size_t)r * IN_C + t * 16 + col] = acc0[v];
        h[(size_t)r * IN_C + (t + 1) * 16 + col] = acc1[v];
      }
    }
  }
}

// ---------------------------------------------------------------------------
// K6: agg init with self-loop contribution: agg[i] = dinv[i]^2 * h[i]
// One float4 per thread.
// ---------------------------------------------------------------------------
__global__ void agg_init_kernel(const float* __restrict__ h,
                                const float* __restrict__ dinv,
                                float* __restrict__ agg, int n) {
  const int idx = blockIdx.x * 256 + threadIdx.x;  // one float4
  const int node = idx >> 5;
  const int q = idx & 31;
  if (node < n) {
    float s = dinv[node];
    s = s * s;
    const float4 v = ((const float4*)(h + (size_t)node * IN_C))[q];
    float4 o;
    o.x = s * v.x; o.y = s * v.y; o.z = s * v.z; o.w = s * v.w;
    ((float4*)(agg + (size_t)node * IN_C))[q] = o;
  }
}

// ---------------------------------------------------------------------------
// K7: edge scatter: agg[col] += dinv[row]*w*dinv[col] * h[row]
// One wave per edge: lane handles 4 channels (float4 gather, 4 f32 atomics).
// h and agg both live in the 192 MB L2, so this runs at L2 atomic rate.
// ---------------------------------------------------------------------------
__global__ void __launch_bounds__(256) edge_scatter_kernel(
    const int* __restrict__ ei, const float* __restrict__ ew,
    const float* __restrict__ dinv, const float* __restrict__ h,
    float* __restrict__ agg, int ne) {
  const int e = blockIdx.x * 8 + (threadIdx.x >> 5);
  if (e >= ne) return;
  const int lane = threadIdx.x & 31;
  const int row = ei[e];        // source
  const int colv = ei[ne + e];  // target
  const float norm = dinv[row] * ew[e] * dinv[colv];
  const float4 v = ((const float4*)(h + (size_t)row * IN_C))[lane];
  float* ap = agg + (size_t)colv * IN_C + lane * 4;
  atomicAdd(ap + 0, norm * v.x);
  atomicAdd(ap + 1, norm * v.y);
  atomicAdd(ap + 2, norm * v.z);
  atomicAdd(ap + 3, norm * v.w);
}

// ---------------------------------------------------------------------------
// K8: out[i] = relu(agg[i]) . cls_w + cls_b   (wave-per-node, shuffle reduce)
// ---------------------------------------------------------------------------
__global__ void __launch_bounds__(256) classify_kernel(
    const float* __restrict__ agg, const float* __restrict__ clw,
    const float* __restrict__ clb, float* __restrict__ out, int n) {
  const int node = blockIdx.x * 8 + (threadIdx.x >> 5);
  if (node >= n) return;
  const int lane = threadIdx.x & 31;
  const float4 a = ((const float4*)(agg + (size_t)node * IN_C))[lane];
  const float4 w = ((const float4*)clw)[lane];
  float s = fmaxf(a.x, 0.0f) * w.x + fmaxf(a.y, 0.0f) * w.y +
            fmaxf(a.z, 0.0f) * w.z + fmaxf(a.w, 0.0f) * w.w;
#pragma unroll
  for (int off = 16; off >= 1; off >>= 1) s += __shfl_down(s, off, 32);
  if (lane == 0) out[node] = s + clb[0];
}

// ---------------------------------------------------------------------------
extern "C" void kernel_launch(void* const* d_in, const int* in_sizes, int n_in,
                              void* d_out, int out_size, void* d_ws, size_t ws_size,
                              hipStream_t stream) {
  const float* x   = (const float*)d_in[0];
  const int*   ei  = (const int*)d_in[1];   // [2, E] flat: row = ei[e], col = ei[E+e]
  const float* ew  = (const float*)d_in[2];
  const float* W0  = (const float*)d_in[3];
  const float* Wih = (const float*)d_in[4];
  const float* Whh = (const float*)d_in[5];
  const float* bih = (const float*)d_in[6];
  const float* bhh = (const float*)d_in[7];
  const float* clw = (const float*)d_in[8];
  const float* clb = (const float*)d_in[9];

  const int n  = in_sizes[0] / IN_C;  // 100000 nodes
  const int ne = in_sizes[2];         // 600000 edges

  // Workspace layout (floats), float4-aligned sections:
  float* ws   = (float*)d_ws;
  float* W    = ws;                                  // 16384
  float* dinv = W + IN_C * IN_C;                     // n (deg -> dinv in place)
  float* h    = dinv + ((n + 3) & ~3);               // n*128
  float* agg  = h + (size_t)n * IN_C;                // n*128
  float* out  = (float*)d_out;

  evolve_W_kernel<<<IN_C, IN_C, 0, stream>>>(W0, Wih, Whh, bih, bhh, W);
  zero_f32_kernel<<<(n + 255) / 256, 256, 0, stream>>>(dinv, n);
  deg_accum_kernel<<<(ne + 255) / 256, 256, 0, stream>>>(ei, ew, dinv, ne);
  dinv_kernel<<<(n + 255) / 256, 256, 0, stream>>>(dinv, n);
  gemm_xW_kernel<<<(n + 127) / 128, 256, IN_C * IN_C * sizeof(float), stream>>>(x, W, h, n);
  agg_init_kernel<<<((n * 32) + 255) / 256, 256, 0, stream>>>(h, dinv, agg, n);
  edge_scatter_kernel<<<(ne + 7) / 8, 256, 0, stream>>>(ei, ew, dinv, h, agg, ne);
  classify_kernel<<<(n + 7) / 8, 256, 0, stream>>>(agg, clw, clb, out, n);
}